// Encoder_6176162971667
// MI455X (gfx1250) — compile-verified
//
#include <hip/hip_runtime.h>

#define C64 64

typedef float v2f __attribute__((ext_vector_type(2)));
typedef float v8f __attribute__((ext_vector_type(8)));

__device__ __forceinline__ float celuf(float x) {
    return x > 0.0f ? x : expm1f(x);
}

__device__ __forceinline__ float cntf(int i, int k) {
    return (i == 0 || i == k - 1) ? 2.0f : 3.0f;
}

// ---------------------------------------------------------------------------
// Initial GCN: h0[b,n,c] = (sum_src norm(src,n) * x[b,src]) * w[c] + bias[c]
// (x has Cin=1, so x@W is rank-1: aggregate the scalar, then outer-product)
// ---------------------------------------------------------------------------
__global__ void init_gcn(const float* __restrict__ x, const float* __restrict__ w,
                         const float* __restrict__ b, float* __restrict__ out,
                         int k, int Bn) {
    int N = k * k;
    int idx = blockIdx.x * blockDim.x + threadIdx.x;
    if (idx >= Bn * N) return;
    int bb = idx / N, n = idx % N;
    int i = n / k, j = n % k;
    float degd = cntf(i, k) * cntf(j, k);
    const float* xb = x + (size_t)bb * N;
    float s = 0.0f;
    for (int di = -1; di <= 1; ++di) {
        int ii = i + di; if (ii < 0 || ii >= k) continue;
        float ci = cntf(ii, k);
        for (int dj = -1; dj <= 1; ++dj) {
            int jj = j + dj; if (jj < 0 || jj >= k) continue;
            float nrm = 1.0f / sqrtf(ci * cntf(jj, k) * degd);
            s += nrm * xb[ii * k + jj];
        }
    }
    float* o = out + (size_t)idx * C64;
    for (int c = 0; c < C64; ++c) o[c] = s * w[c] + b[c];
}

// ---------------------------------------------------------------------------
// Y[M,64] = X[M,64] @ W[64,64] via V_WMMA_F32_16X16X4_F32 (fp32 matrix pipe).
// Block = 4 waves = one 64x64 output tile. Wave w owns rows [16w,16w+16):
// it loads its 32 A-matrix floats ONCE into registers (16 x v2f), then sweeps
// the four 16-column tiles reading only B fragments from LDS -> 256 WMMA ops
// per block against a single pass over X (no cross-wave A duplication).
// A layout: lane(0..15)=row M, v0/v1 = K+0/K+1; lanes 16..31 same rows, K+2/K+3.
// B layout (mirror): lane = col N, VGPR pair = K. C/D: VGPR j -> M = j + 8*(lane>=16).
// ---------------------------------------------------------------------------
__global__ void __launch_bounds__(128) mm64_wmma(const float* __restrict__ X,
                                                 const float* __restrict__ W,
                                                 float* __restrict__ Y) {
    __shared__ float Ws[64 * 64];
    for (int t = threadIdx.x; t < 64 * 64; t += 128) Ws[t] = W[t];
    __syncthreads();

    int lane = threadIdx.x & 31;
    int wave = threadIdx.x >> 5;
    int mbase = blockIdx.x * 64 + wave * 16;     // this wave's 16-row stripe
    int ml   = lane & 15;
    int koff = (lane >> 4) * 2;
    int mhi  = (lane >> 4) * 8;

    // Load this wave's full A stripe (K=64) into registers once.
    const float* xr = X + (size_t)(mbase + ml) * 64;
    v2f areg[16];
#pragma unroll
    for (int s = 0; s < 16; ++s) {
        areg[s].x = xr[s * 4 + koff];
        areg[s].y = xr[s * 4 + koff + 1];
    }

    // Sweep the four 16-column tiles, reusing A from registers.
#pragma unroll
    for (int nt = 0; nt < 4; ++nt) {
        int n = nt * 16 + ml;
        v8f acc = {0.f, 0.f, 0.f, 0.f, 0.f, 0.f, 0.f, 0.f};
#pragma unroll
        for (int s = 0; s < 16; ++s) {
            v2f bm;
            bm.x = Ws[(s * 4 + koff) * 64 + n];
            bm.y = Ws[(s * 4 + koff + 1) * 64 + n];
            acc = __builtin_amdgcn_wmma_f32_16x16x4_f32(false, areg[s], false, bm,
                                                        (short)0, acc, false, false);
        }
#pragma unroll
        for (int j = 0; j < 8; ++j)
            Y[(size_t)(mbase + mhi + j) * 64 + n] = acc[j];
    }
}

// ---------------------------------------------------------------------------
// Fused: stencil aggregation (+bias) -> InstanceNorm over 64 ch -> (+res) -> CELU
// One wave32 per (batch, node); each lane owns channels {lane, lane+32};
// channel mean/var via __shfl_xor wave reductions (no LDS).
// ---------------------------------------------------------------------------
__global__ void __launch_bounds__(256) gcn_agg_in_celu(const float* __restrict__ H,
                                                       const float* __restrict__ bias,
                                                       const float* __restrict__ RES,
                                                       float* __restrict__ Out,
                                                       int k, int Bn) {
    int N = k * k;
    int wg = blockIdx.x * 8 + (threadIdx.x >> 5);
    if (wg >= Bn * N) return;
    int lane = threadIdx.x & 31;
    int bb = wg / N, n = wg % N;
    int i = n / k, j = n % k;
    float degd = cntf(i, k) * cntf(j, k);
    int c0 = lane, c1 = lane + 32;
    float s0 = bias[c0], s1 = bias[c1];
    const float* Hb = H + (size_t)bb * N * 64;
    for (int di = -1; di <= 1; ++di) {
        int ii = i + di; if (ii < 0 || ii >= k) continue;
        float ci = cntf(ii, k);
        for (int dj = -1; dj <= 1; ++dj) {
            int jj = j + dj; if (jj < 0 || jj >= k) continue;
            float nrm = 1.0f / sqrtf(ci * cntf(jj, k) * degd);
            const float* p = Hb + (size_t)(ii * k + jj) * 64;
            s0 += nrm * p[c0];
            s1 += nrm * p[c1];
        }
    }
    float ps = s0 + s1, sq = s0 * s0 + s1 * s1;
    for (int m = 16; m >= 1; m >>= 1) {
        ps += __shfl_xor(ps, m, 32);
        sq += __shfl_xor(sq, m, 32);
    }
    float mean = ps * (1.0f / 64.0f);
    float var  = sq * (1.0f / 64.0f) - mean * mean;
    float inv  = rsqrtf(var + 1e-5f);
    float y0 = (s0 - mean) * inv, y1 = (s1 - mean) * inv;
    if (RES) {
        const float* r = RES + (size_t)wg * 64;
        y0 += r[c0];
        y1 += r[c1];
    }
    float* o = Out + (size_t)wg * 64;
    o[c0] = celuf(y0);
    o[c1] = celuf(y1);
}

// ---------------------------------------------------------------------------
// 2x2 maxpool over (i,j); out node = (i/2)*(k/2) + (j/2). For odd k only the
// first 2*(k/2) rows/cols participate (matches the reference slice).
// ---------------------------------------------------------------------------
__global__ void pool2(const float* __restrict__ In, float* __restrict__ Out,
                      int k, int Bn) {
    int half = k / 2;
    int total = Bn * half * half * 64;
    int idx = blockIdx.x * blockDim.x + threadIdx.x;
    if (idx >= total) return;
    int c = idx & 63;
    int t = idx >> 6;
    int pj = t % half; t /= half;
    int pi = t % half; int bb = t / half;
    const float* ib = In + (size_t)bb * k * k * 64;
    float m = -3.402823466e38f;
    for (int a = 0; a < 2; ++a)
        for (int d = 0; d < 2; ++d)
            m = fmaxf(m, ib[(size_t)((2 * pi + a) * k + (2 * pj + d)) * 64 + c]);
    Out[(size_t)((bb * half + pi) * half + pj) * 64 + c] = m;
}

// ---------------------------------------------------------------------------
// Power iteration (30 iters) for sigma_max(W), W: [OUT, IN] (OUT==256, IN<=2304).
// Replicates the reference's u/v normalization exactly. One 256-thread block.
// ---------------------------------------------------------------------------
__global__ void __launch_bounds__(256) power_iter(const float* __restrict__ W,
                                                  int OUT, int IN,
                                                  float* __restrict__ sig) {
    __shared__ float u[256];
    __shared__ float v[2304];
    __shared__ float red[256];
    int t = threadIdx.x;
    if (t < OUT) u[t] = 1.0f / sqrtf((float)OUT);
    __syncthreads();

    for (int it = 0; it < 30; ++it) {
        // v = W^T u
        for (int j = t; j < IN; j += 256) {
            float a = 0.f;
            for (int i = 0; i < OUT; ++i) a += W[(size_t)i * IN + j] * u[i];
            v[j] = a;
        }
        __syncthreads();
        float p = 0.f;
        for (int j = t; j < IN; j += 256) p += v[j] * v[j];
        red[t] = p; __syncthreads();
        for (int s = 128; s > 0; s >>= 1) { if (t < s) red[t] += red[t + s]; __syncthreads(); }
        float nv = sqrtf(red[0]) + 1e-12f;
        __syncthreads();
        for (int j = t; j < IN; j += 256) v[j] /= nv;
        __syncthreads();
        // u = W v
        if (t < OUT) {
            float a = 0.f;
            for (int j = 0; j < IN; ++j) a += W[(size_t)t * IN + j] * v[j];
            u[t] = a;
        }
        __syncthreads();
        red[t] = (t < OUT) ? u[t] * u[t] : 0.f;
        __syncthreads();
        for (int s = 128; s > 0; s >>= 1) { if (t < s) red[t] += red[t + s]; __syncthreads(); }
        float nu = sqrtf(red[0]) + 1e-12f;
        __syncthreads();
        if (t < OUT) u[t] /= nu;
        __syncthreads();
    }
    // sigma = u . (W v)
    float a = 0.f;
    if (t < OUT) {
        for (int j = 0; j < IN; ++j) a += W[(size_t)t * IN + j] * v[j];
        a *= u[t];
    }
    red[t] = a; __syncthreads();
    for (int s = 128; s > 0; s >>= 1) { if (t < s) red[t] += red[t + s]; __syncthreads(); }
    if (t == 0) sig[0] = red[0];
}

// ---------------------------------------------------------------------------
// y[b,o] = celu( (x[b,:] . W[o,:]) / sigma + bias[o] )
// ---------------------------------------------------------------------------
__global__ void sn_linear_celu(const float* __restrict__ X, const float* __restrict__ W,
                               const float* __restrict__ bias, const float* __restrict__ sig,
                               float* __restrict__ Y, int IN, int OUT, int Bn) {
    int idx = blockIdx.x * blockDim.x + threadIdx.x;
    if (idx >= Bn * OUT) return;
    int bb = idx / OUT, o = idx % OUT;
    float inv = 1.0f / sig[0];
    const float* xr = X + (size_t)bb * IN;
    const float* wr = W + (size_t)o * IN;
    float a = 0.f;
    for (int i = 0; i < IN; ++i) a += xr[i] * wr[i];
    Y[idx] = celuf(a * inv + bias[o]);
}

extern "C" void kernel_launch(void* const* d_in, const int* in_sizes, int n_in,
                              void* d_out, int out_size, void* d_ws, size_t ws_size,
                              hipStream_t stream) {
    const float* x      = (const float*)d_in[0];
    const float* w_init = (const float*)d_in[1];
    const float* b_init = (const float*)d_in[2];
    const float* w11 = (const float*)d_in[3];  const float* b11 = (const float*)d_in[4];
    const float* w12 = (const float*)d_in[5];  const float* b12 = (const float*)d_in[6];
    const float* w21 = (const float*)d_in[7];  const float* b21 = (const float*)d_in[8];
    const float* w22 = (const float*)d_in[9];  const float* b22 = (const float*)d_in[10];
    const float* w31 = (const float*)d_in[11]; const float* b31 = (const float*)d_in[12];
    const float* w32 = (const float*)d_in[13]; const float* b32 = (const float*)d_in[14];
    const float* lw1 = (const float*)d_in[15]; const float* lb1 = (const float*)d_in[16];
    const float* lw2 = (const float*)d_in[17]; const float* lb2 = (const float*)d_in[18];
    float* out = (float*)d_out;

    const int Bn = 64;
    size_t big = (size_t)Bn * 2500 * 64;            // 10.24M floats per buffer
    float* A   = (float*)d_ws;                       // x / pooled output
    float* Bb  = A  + big;                           // GEMM output (pre-aggregation)
    float* Cc  = Bb + big;                           // x1 / x2
    float* L1  = Cc + big;                           // [64,256] hidden
    float* sig = L1 + (size_t)64 * 256;              // 2 sigma scalars

    // Stage 0: initial GCN on the 50x50 grid -> A = [64, 2500, 64]
    {
        int total = Bn * 2500;
        init_gcn<<<(total + 255) / 256, 256, 0, stream>>>(x, w_init, b_init, A, 50, Bn);
    }

    const int ks[3] = {50, 25, 12};
    const float* W1s[3] = {w11, w21, w31}; const float* B1s[3] = {b11, b21, b31};
    const float* W2s[3] = {w12, w22, w32}; const float* B2s[3] = {b12, b22, b32};
    for (int s = 0; s < 3; ++s) {
        int k = ks[s], N = k * k, M = Bn * N;        // M divisible by 64 for all stages
        mm64_wmma<<<M / 64, 128, 0, stream>>>(A, W1s[s], Bb);
        gcn_agg_in_celu<<<(M + 7) / 8, 256, 0, stream>>>(Bb, B1s[s], nullptr, Cc, k, Bn);
        mm64_wmma<<<M / 64, 128, 0, stream>>>(Cc, W2s[s], Bb);
        gcn_agg_in_celu<<<(M + 7) / 8, 256, 0, stream>>>(Bb, B2s[s], A, Cc, k, Bn);
        int half = k / 2;
        int ptotal = Bn * half * half * 64;
        pool2<<<(ptotal + 255) / 256, 256, 0, stream>>>(Cc, A, k, Bn);
    }

    // A now holds [64, 36, 64] == flattened [64, 2304] (contiguous)
    power_iter<<<1, 256, 0, stream>>>(lw1, 256, 2304, sig);
    sn_linear_celu<<<(Bn * 256 + 255) / 256, 256, 0, stream>>>(A, lw1, lb1, sig, L1, 2304, 256, Bn);
    power_iter<<<1, 256, 0, stream>>>(lw2, 256, 256, sig + 1);
    sn_linear_celu<<<(Bn * 256 + 255) / 256, 256, 0, stream>>>(L1, lw2, lb2, sig + 1, out, 256, 256, Bn);
}